// GraphSAGE_67422396612870
// MI455X (gfx1250) — compile-verified
//
#include <hip/hip_runtime.h>
#include <hip/hip_bf16.h>

typedef __attribute__((ext_vector_type(2))) float v2f;
typedef __attribute__((ext_vector_type(8))) float v8f;

#define N_WAVE 32

// ---------------- zero scratch ----------------
__global__ void zero_ws_kernel(float* __restrict__ p, long long n) {
    long long i = (long long)blockIdx.x * blockDim.x + threadIdx.x;
    long long stride = (long long)gridDim.x * blockDim.x;
    for (; i < n; i += stride) p[i] = 0.0f;
}

// ---------------- degree: cnt[dst] += 1 ----------------
__global__ void degree_kernel(const int* __restrict__ ei, float* __restrict__ cnt, int E) {
    int e = blockIdx.x * blockDim.x + threadIdx.x;
    if (e < E) {
        int d = ei[E + e];
        atomicAdd(&cnt[d], 1.0f);
    }
}

// ---------------- scatter x rows (64 f32) into agg1: 16 threads/edge ----------------
__global__ void scatter64_kernel(const int* __restrict__ ei, const float* __restrict__ x,
                                 float* __restrict__ agg, int E) {
    long long t = (long long)blockIdx.x * blockDim.x + threadIdx.x;
    int e = (int)(t >> 4);
    int q = (int)(t & 15);
    if (e < E) {
        int s = ei[e];
        int d = ei[E + e];
        const float4 v = ((const float4*)(x + (size_t)s * 64))[q];
        float* o = agg + (size_t)d * 64 + q * 4;
        atomicAdd(o + 0, v.x);
        atomicAdd(o + 1, v.y);
        atomicAdd(o + 2, v.z);
        atomicAdd(o + 3, v.w);
    }
}

// ---------------- scatter h1 rows (32 f32) into agg2: 8 threads/edge ----------------
__global__ void scatter32_kernel(const int* __restrict__ ei, const float* __restrict__ h,
                                 float* __restrict__ agg, int E) {
    long long t = (long long)blockIdx.x * blockDim.x + threadIdx.x;
    int e = (int)(t >> 3);
    int q = (int)(t & 7);
    if (e < E) {
        int s = ei[e];
        int d = ei[E + e];
        const float4 v = ((const float4*)(h + (size_t)s * 32))[q];
        float* o = agg + (size_t)d * 32 + q * 4;
        atomicAdd(o + 0, v.x);
        atomicAdd(o + 1, v.y);
        atomicAdd(o + 2, v.z);
        atomicAdd(o + 3, v.w);
    }
}

// ---------------- layer 1: h1 = relu((agg1/cnt)@W1l + b1 + x@W1r) ----------------
// One wave per 16-row tile; K=64 (16 steps of V_WMMA_F32_16X16X4_F32);
// two 16-col output tiles (D_HID=32). 64 WMMA per wave.
__global__ void __launch_bounds__(128)
sage_l1_gemm_kernel(const float* __restrict__ x, const float* __restrict__ agg,
                    const float* __restrict__ cnt, const float* __restrict__ W1l,
                    const float* __restrict__ b1, const float* __restrict__ W1r,
                    float* __restrict__ h1, int n) {
    const int lane = threadIdx.x & (N_WAVE - 1);
    const int wave = threadIdx.x >> 5;
    const int base = (blockIdx.x * 4 + wave) * 16;
    if (base >= n) return;                     // wave-uniform: EXEC stays all-1s

    const int col  = lane & 15;
    const int half = lane >> 4;                // 0: K={0,1}  1: K={2,3}
    const int koff = half << 1;
    const int row  = base + col;               // A-operand row for this lane

    const float inv = 1.0f / fmaxf(cnt[row], 1.0f);

    v8f acc0, acc1;
    const float bb0 = b1[col];
    const float bb1 = b1[16 + col];
#pragma unroll
    for (int i = 0; i < 8; ++i) { acc0[i] = bb0; acc1[i] = bb1; }

    const float* __restrict__ xr = x   + (size_t)row * 64;
    const float* __restrict__ ar = agg + (size_t)row * 64;

#pragma unroll
    for (int k0 = 0; k0 < 64; k0 += 4) {
        const int k = k0 + koff;
        v2f aA, aX;
        aA[0] = ar[k] * inv;  aA[1] = ar[k + 1] * inv;
        aX[0] = xr[k];        aX[1] = xr[k + 1];

        v2f bl0, br0, bl1, br1;                // W[k][c], row-major [64][32]
        bl0[0] = W1l[k * 32 + col];        bl0[1] = W1l[(k + 1) * 32 + col];
        br0[0] = W1r[k * 32 + col];        br0[1] = W1r[(k + 1) * 32 + col];
        bl1[0] = W1l[k * 32 + 16 + col];   bl1[1] = W1l[(k + 1) * 32 + 16 + col];
        br1[0] = W1r[k * 32 + 16 + col];   br1[1] = W1r[(k + 1) * 32 + 16 + col];

        acc0 = __builtin_amdgcn_wmma_f32_16x16x4_f32(false, aA, false, bl0, (short)0, acc0, false, false);
        acc0 = __builtin_amdgcn_wmma_f32_16x16x4_f32(false, aX, false, br0, (short)0, acc0, false, false);
        acc1 = __builtin_amdgcn_wmma_f32_16x16x4_f32(false, aA, false, bl1, (short)0, acc1, false, false);
        acc1 = __builtin_amdgcn_wmma_f32_16x16x4_f32(false, aX, false, br1, (short)0, acc1, false, false);
    }

    const int rbase = base + (half << 3);      // D layout: VGPR i -> M = i (+8 upper half)
#pragma unroll
    for (int i = 0; i < 8; ++i) {
        const size_t r = (size_t)(rbase + i);
        h1[r * 32 + col]      = fmaxf(acc0[i], 0.0f);
        h1[r * 32 + 16 + col] = fmaxf(acc1[i], 0.0f);
    }
}

// ---------------- layer 2: out = (agg2/cnt)@W2l + b2 + h1@W2r  (D_OUT=41) ----------------
// K=32 (8 WMMA steps), 3 col tiles (last partially masked). 48 WMMA per wave.
__global__ void __launch_bounds__(128)
sage_l2_gemm_kernel(const float* __restrict__ h1, const float* __restrict__ agg,
                    const float* __restrict__ cnt, const float* __restrict__ W2l,
                    const float* __restrict__ b2, const float* __restrict__ W2r,
                    float* __restrict__ out, int n) {
    const int lane = threadIdx.x & (N_WAVE - 1);
    const int wave = threadIdx.x >> 5;
    const int base = (blockIdx.x * 4 + wave) * 16;
    if (base >= n) return;

    const int col  = lane & 15;
    const int half = lane >> 4;
    const int koff = half << 1;
    const int row  = base + col;

    const float inv = 1.0f / fmaxf(cnt[row], 1.0f);

    v8f acc[3];
#pragma unroll
    for (int ct = 0; ct < 3; ++ct) {
        const int c = ct * 16 + col;
        const float bb = (c < 41) ? b2[c] : 0.0f;
#pragma unroll
        for (int i = 0; i < 8; ++i) acc[ct][i] = bb;
    }

    const float* __restrict__ hr = h1  + (size_t)row * 32;
    const float* __restrict__ ar = agg + (size_t)row * 32;

#pragma unroll
    for (int k0 = 0; k0 < 32; k0 += 4) {
        const int k = k0 + koff;
        v2f aA, aH;
        aA[0] = ar[k] * inv;  aA[1] = ar[k + 1] * inv;
        aH[0] = hr[k];        aH[1] = hr[k + 1];

#pragma unroll
        for (int ct = 0; ct < 3; ++ct) {
            const int c = ct * 16 + col;
            v2f bl, br;                         // W2* row-major [32][41]
            bl[0] = (c < 41) ? W2l[k * 41 + c] : 0.0f;
            bl[1] = (c < 41) ? W2l[(k + 1) * 41 + c] : 0.0f;
            br[0] = (c < 41) ? W2r[k * 41 + c] : 0.0f;
            br[1] = (c < 41) ? W2r[(k + 1) * 41 + c] : 0.0f;
            acc[ct] = __builtin_amdgcn_wmma_f32_16x16x4_f32(false, aA, false, bl, (short)0, acc[ct], false, false);
            acc[ct] = __builtin_amdgcn_wmma_f32_16x16x4_f32(false, aH, false, br, (short)0, acc[ct], false, false);
        }
    }

    const int rbase = base + (half << 3);
#pragma unroll
    for (int ct = 0; ct < 3; ++ct) {
        const int c = ct * 16 + col;
        if (c < 41) {
#pragma unroll
            for (int i = 0; i < 8; ++i) {
                out[(size_t)(rbase + i) * 41 + c] = acc[ct][i];
            }
        }
    }
}

extern "C" void kernel_launch(void* const* d_in, const int* in_sizes, int n_in,
                              void* d_out, int out_size, void* d_ws, size_t ws_size,
                              hipStream_t stream) {
    const float* x   = (const float*)d_in[0];
    const int*   ei  = (const int*)d_in[1];    // edge_index [2, E]: row0=src, row1=dst
    const float* W1l = (const float*)d_in[2];
    const float* b1  = (const float*)d_in[3];
    const float* W1r = (const float*)d_in[4];
    const float* W2l = (const float*)d_in[5];
    const float* b2  = (const float*)d_in[6];
    const float* W2r = (const float*)d_in[7];
    float* out = (float*)d_out;

    const int N = in_sizes[0] / 64;
    const int E = in_sizes[1] / 2;

    // workspace layout (floats): agg1[N*64] | cnt[N] | agg2[N*32] | h1[N*32]
    float* ws   = (float*)d_ws;
    float* agg1 = ws;
    float* cnt  = agg1 + (size_t)N * 64;
    float* agg2 = cnt + N;
    float* h1   = agg2 + (size_t)N * 32;

    // 1) zero accumulators (agg1, cnt, agg2 are contiguous: N*97 floats)
    {
        long long nz = (long long)N * 97;
        zero_ws_kernel<<<2048, 256, 0, stream>>>(ws, nz);
    }
    // 2) degrees
    degree_kernel<<<(E + 255) / 256, 256, 0, stream>>>(ei, cnt, E);
    // 3) scatter x into agg1 (16 threads/edge, float4 each)
    {
        long long t = (long long)E * 16;
        scatter64_kernel<<<(unsigned)((t + 255) / 256), 256, 0, stream>>>(ei, x, agg1, E);
    }
    // 4) layer 1 GEMM + ReLU -> h1
    {
        int tiles = (N + 15) / 16;
        int blocks = (tiles + 3) / 4;
        sage_l1_gemm_kernel<<<blocks, 128, 0, stream>>>(x, agg1, cnt, W1l, b1, W1r, h1, N);
    }
    // 5) scatter h1 into agg2 (8 threads/edge)
    {
        long long t = (long long)E * 8;
        scatter32_kernel<<<(unsigned)((t + 255) / 256), 256, 0, stream>>>(ei, h1, agg2, E);
    }
    // 6) layer 2 GEMM -> out
    {
        int tiles = (N + 15) / 16;
        int blocks = (tiles + 3) / 4;
        sage_l2_gemm_kernel<<<blocks, 128, 0, stream>>>(h1, agg2, cnt, W2l, b2, W2r, out, N);
    }
}